// Logit_Linear_10179072492096
// MI455X (gfx1250) — compile-verified
//
#include <hip/hip_runtime.h>
#include <hip/hip_bf16.h>

#define TILE 2048
#define PASS_THREADS 256

typedef __attribute__((ext_vector_type(4))) unsigned tdm_u32x4;
typedef __attribute__((ext_vector_type(8))) int      tdm_i32x8;
typedef __attribute__((ext_vector_type(4))) int      tdm_i32x4;

// Monotone float<->uint transform: preserves total order as unsigned ints.
__device__ __forceinline__ unsigned fmono(float f) {
    unsigned u = __float_as_uint(f);
    return (u & 0x80000000u) ? ~u : (u | 0x80000000u);
}
__device__ __forceinline__ float funmono(unsigned u) {
    return __uint_as_float((u & 0x80000000u) ? (u ^ 0x80000000u) : ~u);
}

// Workspace word offsets (C = number of classes).
__host__ __device__ __forceinline__ int off_max()          { return 0; }
__host__ __device__ __forceinline__ int off_prefix(int)    { return 8; }
__host__ __device__ __forceinline__ int off_pmask(int C)   { return 8 + C; }
__host__ __device__ __forceinline__ int off_k(int C)       { return 8 + 2 * C; }
__host__ __device__ __forceinline__ int off_done(int C)    { return 8 + 3 * C; }
__host__ __device__ __forceinline__ int off_hist(int C)    { return 8 + 4 * C; }
__host__ __device__ __forceinline__ int ws_words(int C)    { return 8 + 4 * C + 256 * C; }

// Issue one 1-D TDM copy: nelem 4-byte elements, global -> LDS (async, TENSORcnt).
// D# packed per CDNA5 ISA 8.3/8.4: group0 {count=1, lds_addr, global_addr, type=2},
// group1 {data_size=4B, tensor_dim0=nelem, tile_dim0=nelem}.
__device__ __forceinline__ void tdm_load_1d_to_lds(unsigned lds_addr, const void* gsrc,
                                                   unsigned nelem) {
    unsigned long long ga = (unsigned long long)(uintptr_t)gsrc;
    tdm_u32x4 g0;
    g0[0] = 1u;                                              // count=1, user mode
    g0[1] = lds_addr;                                        // lds_addr (bytes)
    g0[2] = (unsigned)ga;                                    // global_addr[31:0]
    g0[3] = ((unsigned)(ga >> 32) & 0x01FFFFFFu) | (2u << 30); // addr[56:32], type=2
    tdm_i32x8 g1;
    g1[0] = (int)(2u << 16);                                 // data_size=4B, mask=0
    g1[1] = (int)((nelem & 0xFFFFu) << 16);                  // tensor_dim0[15:0]
    g1[2] = (int)((nelem >> 16) & 0xFFFFu);                  // tensor_dim0[31:16]
    g1[3] = (int)((nelem & 0xFFFFu) << 16);                  // tile_dim0 (1-D tile)
    g1[4] = 0;                                               // tile_dim1/2 unused
    g1[5] = (int)nelem;                                      // tensor_dim0_stride lo
    g1[6] = 0;
    g1[7] = 0;
    tdm_i32x4 z4 = {0, 0, 0, 0};
    tdm_i32x8 z8 = {0, 0, 0, 0, 0, 0, 0, 0};
    __builtin_amdgcn_tensor_load_to_lds(g0, g1, z4, z4, z8, 0);
}

__global__ void init_ws_kernel(unsigned* __restrict__ ws, int words) {
    int stride = gridDim.x * blockDim.x;
    for (int i = blockIdx.x * blockDim.x + threadIdx.x; i < words; i += stride)
        ws[i] = 0u;
}

__global__ void gmax_kernel(const float* __restrict__ score, int n,
                            unsigned* __restrict__ ws) {
    unsigned m = 0u;  // 0 == monotone code of the smallest possible key
    int stride = gridDim.x * blockDim.x;
    for (int i = blockIdx.x * blockDim.x + threadIdx.x; i < n; i += stride)
        m = max(m, fmono(score[i]));
    #pragma unroll
    for (int off = 16; off > 0; off >>= 1) {
        unsigned o = (unsigned)__shfl_xor((int)m, off, 32);
        m = m > o ? m : o;
    }
    if ((threadIdx.x & 31) == 0) atomicMax(ws + off_max(), m);
}

// One radix pass: stream (label, score) tiles through LDS via double-buffered
// TDM DMA, histogram byte (u>>shift)&255 per class for items matching the
// class's current prefix.
__global__ void pass_kernel(const float* __restrict__ score,
                            const int* __restrict__ label,
                            unsigned* __restrict__ ws, int C, int pass, int n) {
    const int shift = 24 - 8 * pass;
    __shared__ int   lab_tile[2][TILE];
    __shared__ float sc_tile[2][TILE];
    __shared__ unsigned spfx[256], spmask[256];

    unsigned* __restrict__ hist = ws + off_hist(C);
    for (int i = threadIdx.x; i < C && i < 256; i += blockDim.x) {
        spfx[i]   = ws[off_prefix(C) + i];
        spmask[i] = ws[off_pmask(C) + i];
    }
    __syncthreads();

    const int tiles_total = (n + TILE - 1) / TILE;
    const int tpb = (tiles_total + gridDim.x - 1) / gridDim.x;
    const int t0 = blockIdx.x * tpb;
    const int t1 = min(t0 + tpb, tiles_total);
    if (t0 >= t1) return;  // uniform per block

    const unsigned lds_lab0 = (unsigned)(uintptr_t)&lab_tile[0][0];
    const unsigned lds_lab1 = (unsigned)(uintptr_t)&lab_tile[1][0];
    const unsigned lds_sc0  = (unsigned)(uintptr_t)&sc_tile[0][0];
    const unsigned lds_sc1  = (unsigned)(uintptr_t)&sc_tile[1][0];

    if (threadIdx.x < 32) {  // wave 0 drives the DMA
        int base = t0 * TILE;
        unsigned len = (unsigned)min(TILE, n - base);
        tdm_load_1d_to_lds(lds_lab0, label + base, len);
        tdm_load_1d_to_lds(lds_sc0,  score + base, len);
    }
    for (int t = t0; t < t1; ++t) {
        const int buf = (t - t0) & 1;
        if (threadIdx.x < 32) {
            if (t + 1 < t1) {
                int nb = (t + 1) * TILE;
                unsigned nl = (unsigned)min(TILE, n - nb);
                tdm_load_1d_to_lds(buf ? lds_lab0 : lds_lab1, label + nb, nl);
                tdm_load_1d_to_lds(buf ? lds_sc0 : lds_sc1,   score + nb, nl);
                __builtin_amdgcn_s_wait_tensorcnt(2);  // current tile's pair done
            } else {
                __builtin_amdgcn_s_wait_tensorcnt(0);
            }
            __asm__ volatile("" ::: "memory");
        }
        __syncthreads();
        const int base = t * TILE;
        const int len = min(TILE, n - base);
        for (int j = threadIdx.x; j < len; j += blockDim.x) {
            const int c = lab_tile[buf][j];
            const unsigned u = fmono(sc_tile[buf][j]);
            unsigned pm, pf;
            if (c < 256) { pm = spmask[c]; pf = spfx[c]; }
            else         { pm = ws[off_pmask(C) + c]; pf = ws[off_prefix(C) + c]; }
            if ((u & pm) == pf)
                atomicAdd(&hist[(unsigned)c * 256u + ((u >> shift) & 255u)], 1u);
        }
        __syncthreads();  // tile buffer may be overwritten next iteration
    }
}

// One block per class: pick the bucket containing the k-th smallest, update
// prefix/pmask/k, reset this class's histogram, emit result on the last pass.
__global__ void pick_kernel(unsigned* __restrict__ ws,
                            const int* __restrict__ epsp,
                            float* __restrict__ out, int C, int pass) {
    const int c = blockIdx.x;
    const int t = threadIdx.x;
    unsigned* __restrict__ hist = ws + off_hist(C) + (unsigned)c * 256u;
    __shared__ int h[256];
    h[t] = (int)hist[t];
    hist[t] = 0u;  // clean for next pass
    __syncthreads();
    if (t != 0) return;
    if (ws[off_done(C) + c] != 0u) return;

    const int shift = 24 - 8 * pass;
    int k;
    if (pass == 0) {
        int total = 0;
        for (int b = 0; b < 256; ++b) total += h[b];
        // Match reference: idx = int32(float32(count) * (1 - eps)), truncation.
        const float frac = (float)(1.0 - (double)epsp[0] / 100.0);
        k = (int)((float)total * frac);
        if (total == 0 || k >= total) {  // quantile falls in the max_ele fill
            out[c] = funmono(ws[off_max()]);
            ws[off_done(C) + c] = 1u;
            return;
        }
    } else {
        k = (int)ws[off_k(C) + c];
    }
    int run = 0, b = 0;
    for (; b < 256; ++b) {
        const int hb = h[b];
        if (run + hb > k) break;
        run += hb;
    }
    ws[off_k(C) + c] = (unsigned)(k - run);
    const unsigned np = ws[off_prefix(C) + c] | ((unsigned)b << shift);
    ws[off_prefix(C) + c] = np;
    ws[off_pmask(C) + c] |= (0xFFu << shift);
    if (pass == 3) out[c] = funmono(np);  // prefix == exact f32 bit pattern
}

extern "C" void kernel_launch(void* const* d_in, const int* in_sizes, int n_in,
                              void* d_out, int out_size, void* d_ws, size_t ws_size,
                              hipStream_t stream) {
    const float* score = (const float*)d_in[0];
    const int*   label = (const int*)d_in[1];
    // d_in[2] = class_num (== out_size, used from host side)
    const int*   eps   = (const int*)d_in[3];
    float*    out = (float*)d_out;
    unsigned* ws  = (unsigned*)d_ws;
    const int n = in_sizes[0];
    const int C = out_size;

    const int words = ws_words(C);
    int zb = (words + 255) / 256; if (zb > 256) zb = 256;
    init_ws_kernel<<<zb, 256, 0, stream>>>(ws, words);

    int gb = (n + 255) / 256; if (gb > 1024) gb = 1024; if (gb < 1) gb = 1;
    gmax_kernel<<<gb, 256, 0, stream>>>(score, n, ws);

    int tiles = (n + TILE - 1) / TILE;
    int pb = tiles; if (pb > 1024) pb = 1024; if (pb < 1) pb = 1;
    for (int p = 0; p < 4; ++p) {
        pass_kernel<<<pb, PASS_THREADS, 0, stream>>>(score, label, ws, C, p, n);
        pick_kernel<<<C, 256, 0, stream>>>(ws, eps, out, C, p);
    }
}